// NeuromorphicCore_75969381532094
// MI455X (gfx1250) — compile-verified
//
#include <hip/hip_runtime.h>
#include <hip/hip_bf16.h>

#define NM_D 8192
#define NM_T 100
#define NM_THRESH 0.5f

// Per-wave column tile = 16, waves/block = 8 -> 128 columns per block.
// K-slice per block row = 1024.
#define NM_KSLICE 1024
#define NM_COLS_PER_BLOCK 128

typedef __attribute__((ext_vector_type(2))) float v2f;
typedef __attribute__((ext_vector_type(8))) float v8f;

// ---------------------------------------------------------------------------
// Zero the workspace accumulators: ws[0..D-1] = drive, ws[D] = sum|W|.
// Must run every launch (harness does not re-poison between replays).
// ---------------------------------------------------------------------------
__global__ void nm_init_kernel(float* __restrict__ ws) {
    int i = blockIdx.x * blockDim.x + threadIdx.x;
    if (i <= NM_D) ws[i] = 0.0f;
}

// ---------------------------------------------------------------------------
// Fused: drive[j] += sum_k x[k] * W[k][j]  (via V_WMMA_F32_16X16X4_F32)
//        abssum  += sum |W|                (each W element read exactly once)
//
// Wave w of block (bx,by) owns columns [bx*128 + w*16, +16) and K rows
// [by*1024, +1024). A-matrix rows are all broadcast copies of the 4-wide
// x chunk, so every row of D equals the drive tile; row M=0 lives in c[0]
// on lanes 0..15.
// ---------------------------------------------------------------------------
__global__ void __launch_bounds__(256)
nm_matvec_kernel(const float* __restrict__ x, const float* __restrict__ W,
                 float* __restrict__ drive, float* __restrict__ abssum) {
    const int lane = threadIdx.x & 31;
    const int wave = threadIdx.x >> 5;
    const int j0   = blockIdx.x * NM_COLS_PER_BLOCK + wave * 16; // column tile base
    const int k0   = blockIdx.y * NM_KSLICE;                     // K slice base
    const int half = lane >> 4;       // 0: lanes 0-15 (K=0,1) ; 1: lanes 16-31 (K=2,3)
    const int n    = lane & 15;       // column within tile
    const int ka   = half * 2;        // first K handled by this half-wave

    v8f c = {};
    float asum = 0.0f;

    #pragma unroll 2
    for (int k = 0; k < NM_KSLICE; k += 4) {
        const int kk = k0 + k;

        // A (16x4 f32): every row M holds the same x chunk.
        // lanes 0-15: VGPR0=K0, VGPR1=K1 ; lanes 16-31: VGPR0=K2, VGPR1=K3
        v2f a;
        a.x = x[kk + ka];
        a.y = x[kk + ka + 1];

        // B (4x16 f32): rows K=ka, ka+1 of W, columns j0..j0+15 on lanes.
        const float* wp = W + (size_t)(kk + ka) * NM_D + (j0 + n);
        v2f b;
        b.x = wp[0];
        b.y = wp[NM_D];

        c = __builtin_amdgcn_wmma_f32_16x16x4_f32(
                /*neg_a=*/false, a, /*neg_b=*/false, b,
                /*c_mod=*/(short)0, c, /*reuse_a=*/false, /*reuse_b=*/false);

        // Each W element is touched exactly once across the whole grid.
        asum += fabsf(b.x) + fabsf(b.y);
    }

    // wave32 reduction of the |W| partial sum
    #pragma unroll
    for (int m = 16; m >= 1; m >>= 1)
        asum += __shfl_xor(asum, m, 32);
    if (lane == 0)
        atomicAdd(abssum, asum);

    // D row M=0 (the drive tile) is c[0] on lanes 0..15.
    if (lane < 16)
        atomicAdd(&drive[j0 + lane], c[0]);
}

// ---------------------------------------------------------------------------
// 100-step leaky-integrate-and-fire scan, one thread per neuron.
// out layout (flat, return order): spikes[T,D] | potentials[T,D] | scalar
// ---------------------------------------------------------------------------
__global__ void __launch_bounds__(256)
nm_scan_kernel(const float* __restrict__ drive, const float* __restrict__ pot0,
               const float* __restrict__ abssum, float* __restrict__ out) {
    const int j = blockIdx.x * blockDim.x + threadIdx.x;
    if (j < NM_D) {
        const float d = drive[j];
        float pot = pot0[j];
        float* spikes = out;
        float* pots   = out + (size_t)NM_T * NM_D;
        #pragma unroll 4
        for (int t = 0; t < NM_T; ++t) {
            pot += d;
            float spike = (pot >= NM_THRESH) ? 1.0f : 0.0f;
            pot *= (1.0f - spike);
            spikes[(size_t)t * NM_D + j] = spike;
            pots[(size_t)t * NM_D + j]   = pot;
        }
    }
    if (blockIdx.x == 0 && threadIdx.x == 0) {
        out[(size_t)2 * NM_T * NM_D] =
            abssum[0] * (1.0f / ((float)NM_D * (float)NM_D));
    }
}

extern "C" void kernel_launch(void* const* d_in, const int* in_sizes, int n_in,
                              void* d_out, int out_size, void* d_ws, size_t ws_size,
                              hipStream_t stream) {
    const float* x    = (const float*)d_in[0]; // [D]
    const float* W    = (const float*)d_in[1]; // [D,D] row-major
    const float* pot0 = (const float*)d_in[2]; // [D] (zeros)
    float* out = (float*)d_out;
    float* ws  = (float*)d_ws;                 // ws[0..D-1]=drive, ws[D]=abssum

    (void)in_sizes; (void)n_in; (void)out_size; (void)ws_size;

    // 1) zero accumulators (D+1 floats)
    nm_init_kernel<<<(NM_D + 1 + 255) / 256, 256, 0, stream>>>(ws);

    // 2) fused WMMA matvec + |W| sum: 64 column-blocks x 8 K-slices
    dim3 grid(NM_D / NM_COLS_PER_BLOCK, NM_D / NM_KSLICE);
    nm_matvec_kernel<<<grid, 256, 0, stream>>>(x, W, ws, ws + NM_D);

    // 3) LIF scan + scalar output
    nm_scan_kernel<<<NM_D / 256, 256, 0, stream>>>(ws, pot0, ws + NM_D, out);
}